// PointToDense_79310866088270
// MI455X (gfx1250) — compile-verified
//
#include <hip/hip_runtime.h>
#include <cstdint>
#include <cstddef>

// ---------------------------------------------------------------------------
// Types for CDNA5 WMMA f32 16x16x4:  A = 16x4 (2 VGPR), B = 4x16 (2 VGPR),
// C/D = 16x16 f32 (8 VGPR).  Wave32.
// ---------------------------------------------------------------------------
typedef __attribute__((ext_vector_type(2))) float v2f;
typedef __attribute__((ext_vector_type(8))) float v8f;
typedef __attribute__((ext_vector_type(4))) unsigned int u32x4;
typedef __attribute__((ext_vector_type(8))) int i32x8;
typedef __attribute__((ext_vector_type(4))) int i32x4;

#define SMAXCH 544            // max channel count (conv3 input) for stats bufs
#define NEGINF_KEY ((int)0x807FFFFF)   // order-preserving encoding of -inf

#if defined(__gfx1250__) && __has_builtin(__builtin_amdgcn_tensor_load_to_lds) && __has_builtin(__builtin_amdgcn_s_wait_tensorcnt)
#define USE_TDM 1
#else
#define USE_TDM 0
#endif

__device__ __forceinline__ float gelu_f(float x) {
  // jax.nn.gelu default (approximate=True, tanh form)
  const float k0 = 0.7978845608028654f;
  float t = tanhf(k0 * (x + 0.044715f * x * x * x));
  return 0.5f * x * (1.0f + t);
}
__device__ __forceinline__ float silu_f(float x) {
  return x / (1.0f + expf(-x));
}
// monotonic float <-> int key mapping so atomicMax(int) == float max
__device__ __forceinline__ int f2key(float f) {
  int i = __float_as_int(f);
  return i >= 0 ? i : (i ^ 0x7fffffff);
}
__device__ __forceinline__ float key2f(int k) {
  int i = k >= 0 ? k : (k ^ 0x7fffffff);
  return __int_as_float(i);
}

#if USE_TDM
// Issue one TDM 2D tile load: global (elements of 4B) -> LDS, with one pad
// dword inserted after every 32 stored dwords (LDS row stride 33).
// dim0/dim1 are tile-relative remaining tensor extents (rows/cols past them
// read as zero -> automatic zero padding of edge tiles).
__device__ __forceinline__ void tdm_load_2d(const float* gbase, unsigned ldsOff,
                                            int dim0, int dim1, int stride0,
                                            int tile0, int tile1) {
  unsigned long long ga = (unsigned long long)(uintptr_t)gbase;
  u32x4 g0;
  g0[0] = 1u;                                      // count=1 (valid descriptor)
  g0[1] = ldsOff;                                  // lds_addr (bytes)
  g0[2] = (unsigned)(ga & 0xffffffffu);            // global_addr[31:0]
  g0[3] = (unsigned)((ga >> 32) & 0x01ffffffu) | (2u << 30);  // addr[56:32] | type=2
  i32x8 g1;
  g1[0] = (2 << 16) | (1 << 20) | (4 << 22);       // data_size=4B, pad_en, every 32 dw, +1 dw
  g1[1] = (int)(((unsigned)dim0 & 0xffffu) << 16); // tensor_dim0[15:0]
  g1[2] = (int)((((unsigned)dim0 >> 16) & 0xffffu) | (((unsigned)dim1 & 0xffffu) << 16));
  g1[3] = (int)((((unsigned)dim1 >> 16) & 0xffffu) | ((unsigned)tile0 << 16));
  g1[4] = tile1 & 0xffff;                          // tile_dim1 (tile_dim2 = 0)
  g1[5] = stride0;                                 // tensor_dim0_stride[31:0]
  g1[6] = 0;
  g1[7] = 0;
  i32x4 z4 = {0, 0, 0, 0};
  i32x8 z8 = {0, 0, 0, 0, 0, 0, 0, 0};
  __builtin_amdgcn_tensor_load_to_lds(g0, g1, z4, z4, z8, 0);
}
#endif

// ---------------------------------------------------------------------------
// Small utility kernels
// ---------------------------------------------------------------------------
__global__ void k_fill_f32(float* p, float v, int n) {
  int i = blockIdx.x * 256 + threadIdx.x;
  if (i < n) p[i] = v;
}
__global__ void k_fill_i32(int* p, int v, int n) {
  int i = blockIdx.x * 256 + threadIdx.x;
  if (i < n) p[i] = v;
}
__global__ void k_decode_max(float* p, int n) {   // int keys -> floats, in place
  int i = blockIdx.x * 256 + threadIdx.x;
  if (i < n) p[i] = key2f(__float_as_int(p[i]));
}
// xy (B*N, 2) -> xT (2, B*N)
__global__ void k_transpose_xy(const float* __restrict__ xy, float* __restrict__ xT, int total) {
  int i = blockIdx.x * 256 + threadIdx.x;
  if (i < total) {
    xT[i]         = xy[2 * i + 0];
    xT[total + i] = xy[2 * i + 1];
  }
}

// ---------------------------------------------------------------------------
// Brute-force KNN (top-10 smallest squared distance, self included, stable
// ties by lower index -- matches lax.top_k on -d2).  One thread per query
// point; candidate points tiled through LDS.   grid = (N/256, B)
// ---------------------------------------------------------------------------
template <int C>
__global__ void __launch_bounds__(256)
k_knn(const float* __restrict__ X, int stride, int* __restrict__ idx, int N) {
  const int KNN = 10;
  int b = blockIdx.y;
  int n = blockIdx.x * 256 + threadIdx.x;
  int base = b * N;

  float p[C];
#pragma unroll
  for (int c = 0; c < C; ++c) p[c] = X[c * stride + base + n];

  float bd[KNN];
  int   bi[KNN];
#pragma unroll
  for (int k = 0; k < KNN; ++k) { bd[k] = 3.4e38f; bi[k] = 0; }

  __shared__ float Q[C][128];
  for (int m0 = 0; m0 < N; m0 += 128) {
    __syncthreads();
    for (int e = threadIdx.x; e < C * 128; e += 256) {
      int c = e >> 7, mm = e & 127;
      Q[c][mm] = X[c * stride + base + m0 + mm];
    }
    __syncthreads();
    for (int mm = 0; mm < 128; ++mm) {
      float d = 0.f;
#pragma unroll
      for (int c = 0; c < C; ++c) { float df = p[c] - Q[c][mm]; d += df * df; }
      if (d < bd[KNN - 1]) {
        bd[KNN - 1] = d; bi[KNN - 1] = m0 + mm;
#pragma unroll
        for (int k = KNN - 1; k > 0; --k) {
          if (bd[k] < bd[k - 1]) {
            float td = bd[k]; bd[k] = bd[k - 1]; bd[k - 1] = td;
            int   ti = bi[k]; bi[k] = bi[k - 1]; bi[k - 1] = ti;
          }
        }
      }
    }
  }
#pragma unroll
  for (int k = 0; k < KNN; ++k) idx[(base + n) * KNN + k] = bi[k];
}

// ---------------------------------------------------------------------------
// Graph feature: F[0..C) = x[:, idx] - x[:, n];  F[C..2C) = x[:, n]
// over expanded columns col = (b*N + n)*10 + k
// ---------------------------------------------------------------------------
template <int C>
__global__ void k_feature(const float* __restrict__ X, int stride,
                          const int* __restrict__ idx, float* __restrict__ F,
                          int colsE, int N) {
  int col = blockIdx.x * 256 + threadIdx.x;
  if (col >= colsE) return;
  int bn = col / 10;
  int b  = bn / N;
  int nei = idx[col];
  int neicol = b * N + nei;
#pragma unroll
  for (int c = 0; c < C; ++c) {
    float cen = X[c * stride + bn];
    float nv  = X[c * stride + neicol];
    F[c * colsE + col]       = nv - cen;
    F[(c + C) * colsE + col] = cen;
  }
}

// ---------------------------------------------------------------------------
// Generic WMMA f32 GEMM:  Z[o, col] = sum_c W[o,c] * X[c,col] + bias[o]
// Tile: 64 (O) x 32 (cols), 8 waves, K in chunks of 32.
// Staging: double-buffered TDM (tensor_load_to_lds) when available --
// wave 0 DMAs next chunk's W (64x32) and X (32x32) tiles into LDS (stride-33
// rows via the TDM pad feature, edge tiles auto zero-filled) while all waves
// run 8 WMMAs on the current chunk.  One s_wait_tensorcnt + barrier per chunk.
// mode bit0: store Z   bit1: per-channel sum/sumsq stats   bit2: BN+GELU+max
// ---------------------------------------------------------------------------
__global__ void __launch_bounds__(256)
k_gemm(const float* __restrict__ W, const float* __restrict__ bias,
       const float* __restrict__ X, int O, int C, int COLS,
       float* __restrict__ Zout, float* __restrict__ sums,
       const float* __restrict__ mean, const float* __restrict__ rstd,
       const float* __restrict__ gamma, const float* __restrict__ beta,
       int* __restrict__ maxout, int maxStride, int kdiv, int mode) {
  __shared__ float Wl[2][64][33];   // [buf][o in tile][c in chunk]
  __shared__ float Xl[2][32][33];   // [buf][c in chunk][col in tile]

  int t = threadIdx.x;
  int lane = t & 31, w = t >> 5;
  int half = lane >> 4, l15 = lane & 15;
  int o0  = blockIdx.x * 64;
  int n0g = blockIdx.y * 32;
  int m0  = (w & 3) * 16;        // wave's O sub-tile
  int nn0 = (w >> 2) * 16;       // wave's col sub-tile

  v8f acc = {};
  int nch = (C + 31) >> 5;

#if USE_TDM
  if (w == 0) {
    tdm_load_2d(W + o0 * C, (unsigned)(uintptr_t)&Wl[0][0][0], C, O - o0, C, 32, 64);
    tdm_load_2d(X + n0g,    (unsigned)(uintptr_t)&Xl[0][0][0], COLS - n0g, C, COLS, 32, 32);
  }
  for (int ch = 0; ch < nch; ++ch) {
    int cur = ch & 1;
    __builtin_amdgcn_s_wait_tensorcnt((short)0);
    __syncthreads();
    if (w == 0 && ch + 1 < nch) {
      int c0n = (ch + 1) << 5;
      tdm_load_2d(W + o0 * C + c0n,    (unsigned)(uintptr_t)&Wl[cur ^ 1][0][0],
                  C - c0n, O - o0, C, 32, 64);
      tdm_load_2d(X + c0n * COLS + n0g, (unsigned)(uintptr_t)&Xl[cur ^ 1][0][0],
                  COLS - n0g, C - c0n, COLS, 32, 32);
    }
#pragma unroll
    for (int kk = 0; kk < 8; ++kk) {
      int kb = kk * 4 + 2 * half;
      v2f a, b;
      a.x = Wl[cur][m0 + l15][kb];
      a.y = Wl[cur][m0 + l15][kb + 1];
      b.x = Xl[cur][kb][nn0 + l15];
      b.y = Xl[cur][kb + 1][nn0 + l15];
      acc = __builtin_amdgcn_wmma_f32_16x16x4_f32(false, a, false, b,
                                                  (short)0, acc, false, false);
    }
  }
#else
  // fallback: manual branchless staging (clamped addresses + selects)
  int wr = t >> 2;               // W row in tile   (0..63)
  int wcb = (t & 3) * 8;         // W col base      (0,8,16,24)
  int xc_ = t >> 3;              // X channel       (0..31)
  int xcolb = (t & 7) * 4;       // X col base      (0,4,...,28)
  int wrow = min(o0 + wr, O - 1);
  for (int ch = 0; ch < nch; ++ch) {
    int c0 = ch << 5;
    {
      int cn = min(c0 + 32 + xc_, C - 1);
      __builtin_prefetch(&X[cn * COLS + n0g + xcolb], 0, 1);
    }
#pragma unroll
    for (int j = 0; j < 8; ++j) {
      int c = wcb + j;
      int cc = min(c0 + c, C - 1);
      float v = W[wrow * C + cc];
      v = (o0 + wr < O && c0 + c < C) ? v : 0.f;
      Wl[0][wr][c] = v;
    }
    {
      int cc = min(c0 + xc_, C - 1);
      const float4 xv = *reinterpret_cast<const float4*>(&X[cc * COLS + n0g + xcolb]);
      bool ok = (c0 + xc_ < C);
      Xl[0][xc_][xcolb + 0] = ok ? xv.x : 0.f;
      Xl[0][xc_][xcolb + 1] = ok ? xv.y : 0.f;
      Xl[0][xc_][xcolb + 2] = ok ? xv.z : 0.f;
      Xl[0][xc_][xcolb + 3] = ok ? xv.w : 0.f;
    }
    __syncthreads();
#pragma unroll
    for (int kk = 0; kk < 8; ++kk) {
      int kb = kk * 4 + 2 * half;
      v2f a, b;
      a.x = Wl[0][m0 + l15][kb];
      a.y = Wl[0][m0 + l15][kb + 1];
      b.x = Xl[0][kb][nn0 + l15];
      b.y = Xl[0][kb + 1][nn0 + l15];
      acc = __builtin_amdgcn_wmma_f32_16x16x4_f32(false, a, false, b,
                                                  (short)0, acc, false, false);
    }
    __syncthreads();
  }
#endif

  // epilogue -- C/D layout: vgpr v, lane: M = v + 8*half, N = l15
#pragma unroll
  for (int v = 0; v < 8; ++v) {
    int o = o0 + m0 + v + 8 * half;
    int col = n0g + nn0 + l15;
    bool valid = (o < O);
    int oc = valid ? o : 0;
    float bz = valid ? (acc[v] + (bias ? bias[oc] : 0.f)) : 0.f;
    if (mode & 1) {
      if (valid) Zout[o * COLS + col] = bz;
    }
    if (mode & 2) {
      float s = bz, s2 = bz * bz;
      for (int off = 1; off < 16; off <<= 1) {
        s  += __shfl_xor(s, off, 32);
        s2 += __shfl_xor(s2, off, 32);
      }
      if (l15 == 0 && valid) {
        atomicAdd(&sums[o], s);
        atomicAdd(&sums[o + SMAXCH], s2);
      }
    }
    if (mode & 4) {
      if (valid) {
        float u = (bz - mean[o]) * rstd[o] * gamma[o] + beta[o];
        u = gelu_f(u);
        atomicMax(&maxout[o * maxStride + col / kdiv], f2key(u));
      }
    }
  }
}

// sums -> mean, rstd
__global__ void k_finalize_stats(const float* __restrict__ sums, float* __restrict__ mean,
                                 float* __restrict__ rstd, int O, float invCnt) {
  int o = blockIdx.x * 256 + threadIdx.x;
  if (o < O) {
    float m  = sums[o] * invCnt;
    float vv = sums[o + SMAXCH] * invCnt - m * m;
    mean[o] = m;
    rstd[o] = rsqrtf(vv + 1e-5f);
  }
}

// in-place BN + GELU, grid = (ceil(COLS/256), O)
__global__ void k_bngelu(float* __restrict__ Z, const float* __restrict__ mean,
                         const float* __restrict__ rstd, const float* __restrict__ g,
                         const float* __restrict__ bt, int COLS) {
  int o = blockIdx.y;
  int col = blockIdx.x * 256 + threadIdx.x;
  if (col < COLS) {
    int i = o * COLS + col;
    Z[i] = gelu_f((Z[i] - mean[o]) * rstd[o] * g[o] + bt[o]);
  }
}

// time MLP: t[b,o] = sum_c gelu(temb[b,c]) * tW[o,c] + tb[o]
__global__ void k_time(const float* __restrict__ temb, const float* __restrict__ tW,
                       const float* __restrict__ tb, float* __restrict__ t,
                       int B, int Oc, int Cc) {
  int i = blockIdx.x * 256 + threadIdx.x;
  if (i >= B * Oc) return;
  int b = i / Oc, o = i - b * Oc;
  float s = tb[o];
  for (int c = 0; c < Cc; ++c) s += gelu_f(temb[b * Cc + c]) * tW[o * Cc + c];
  t[i] = s;
}

// RMSNorm over channels + optional (sc+1)/sh modulation + SiLU + optional
// residual add.  Block handles 16 columns; 16 thread-groups over channels.
// toOut: write in (b, c, n) batch-major layout (final output).
__global__ void __launch_bounds__(256)
k_blocknorm(const float* __restrict__ Hin, float* __restrict__ Hout,
            const float* __restrict__ g, const float* __restrict__ tbuf,
            const float* __restrict__ res, int useMod, int CH, int COLS,
            int N, int toOut) {
  __shared__ float part[16][16];
  __shared__ float nrm[16];
  int col0 = blockIdx.x * 16;
  int tc = threadIdx.x & 15;     // column
  int tg = threadIdx.x >> 4;     // channel group
  int col = col0 + tc;
  int cper = CH / 16;

  float s = 0.f;
  for (int j = 0; j < cper; ++j) {
    int c = tg * cper + j;
    float v = Hin[c * COLS + col];
    s += v * v;
  }
  part[tg][tc] = s;
  __syncthreads();
  if (tg == 0) {
    float tot = 0.f;
#pragma unroll
    for (int j = 0; j < 16; ++j) tot += part[j][tc];
    float n = sqrtf(tot);
    nrm[tc] = (1.f / fmaxf(n, 1e-12f)) * sqrtf((float)CH);
  }
  __syncthreads();
  float scale = nrm[tc];
  int b = col / N;
  for (int j = 0; j < cper; ++j) {
    int c = tg * cper + j;
    float v = Hin[c * COLS + col] * scale * g[c];
    if (useMod) {
      float sc = tbuf[b * (2 * CH) + c];
      float sh = tbuf[b * (2 * CH) + CH + c];
      v = v * (sc + 1.f) + sh;
    }
    v = silu_f(v);
    if (res) v += res[c * COLS + col];
    if (toOut) Hout[(b * CH + c) * N + (col - b * N)] = v;
    else       Hout[c * COLS + col] = v;
  }
}

// ---------------------------------------------------------------------------
// Host-side orchestration
// ---------------------------------------------------------------------------
static inline size_t alup(size_t x) { return (x + 255) & ~(size_t)255; }

extern "C" void kernel_launch(void* const* d_in, const int* in_sizes, int n_in,
                              void* d_out, int out_size, void* d_ws, size_t ws_size,
                              hipStream_t stream) {
  (void)out_size; (void)ws_size;
  const int B = 16, N = 2048, NT = B * N;       // 32768
  const int KNN = 10, CE = NT * KNN;            // 327680

  // --- resolve input leaves by size (params assumed jax sorted-key flatten) ---
  int i_xy = -1, i_temb = -1;
  int c1w0 = -1, c1w1 = -1, c2w0 = -1, c2w1 = -1, c3w0 = -1, c3w1 = -1;
  int big[8]; int nbig = 0;
  for (int i = 0; i < n_in; ++i) {
    int s = in_sizes[i];
    if      (s == 65536)  i_xy = i;
    else if (s == 4096)   i_temb = i;
    else if (s == 44)     c1w0 = i;
    else if (s == 352)    c1w1 = i;
    else if (s == 11584)  c2w0 = i;
    else if (s == 92672)  c2w1 = i;
    else if (s == 286688) c3w0 = i;
    else if (s == 269824) c3w1 = i;
    else if (s == 262144) { if (nbig < 8) big[nbig++] = i; }
  }
  int i_tW = -1, i_b1W = -1, i_b2W = -1;
  for (int j = 0; j < nbig; ++j) {
    int i = big[j];
    if (i + 1 < n_in && in_sizes[i + 1] == 1024) i_tW = i;
    else if (i_b1W < 0) i_b1W = i;
    else i_b2W = i;
  }
  auto F = [&](int i) { return (const float*)d_in[i]; };
  const float* xy   = F(i_xy);
  const float* temb = F(i_temb);
  // conv dicts flatten as W, b, bt, g
  const float *W1a = F(c1w0), *b1a = F(c1w0 + 1), *bt1a = F(c1w0 + 2), *g1a = F(c1w0 + 3);
  const float *W1b = F(c1w1), *b1b = F(c1w1 + 1), *bt1b = F(c1w1 + 2), *g1b = F(c1w1 + 3);
  const float *W2a = F(c2w0), *b2a = F(c2w0 + 1), *bt2a = F(c2w0 + 2), *g2a = F(c2w0 + 3);
  const float *W2b = F(c2w1), *b2b = F(c2w1 + 1), *bt2b = F(c2w1 + 2), *g2b = F(c2w1 + 3);
  const float *W3a = F(c3w0), *b3a = F(c3w0 + 1), *bt3a = F(c3w0 + 2), *g3a = F(c3w0 + 3);
  const float *W3b = F(c3w1), *b3b = F(c3w1 + 1), *bt3b = F(c3w1 + 2), *g3b = F(c3w1 + 3);
  const float *tW  = F(i_tW),  *tb  = F(i_tW + 1);
  const float *B1W = F(i_b1W), *B1b = F(i_b1W + 1), *B1g = F(i_b1W + 2);
  const float *B2W = F(i_b2W), *B2b = F(i_b2W + 1), *B2g = F(i_b2W + 2);

  // --- workspace layout ---
  char* ws = (char*)d_ws;
  size_t off = 0;
  size_t off_sums = off; off += alup(2 * SMAXCH * 4);
  size_t off_bnp  = off; off += alup(2 * SMAXCH * 4);
  size_t off_t    = off; off += alup((size_t)B * 1024 * 4);
  size_t off_idx1 = off; off += alup((size_t)CE * 4);
  size_t off_idx2 = off; off += alup((size_t)CE * 4);
  size_t off_xT   = off; off += alup((size_t)2 * NT * 4);
  size_t off_xc   = off; off += alup((size_t)544 * NT * 4);
  size_t off_x3   = off; off += alup((size_t)512 * NT * 4);
  size_t off_big  = off;
  size_t off_f1  = off_big, off_z1a = off_big + alup((size_t)4  * CE * 4);
  size_t off_f2  = off_big, off_z2a = off_big + alup((size_t)64 * CE * 4);
  size_t off_z3a = off_big;
  size_t off_H   = off_big, off_H2  = off_big + alup((size_t)512 * NT * 4);

  float* sums = (float*)(ws + off_sums);
  float* mean = (float*)(ws + off_bnp);
  float* rstd = mean + SMAXCH;
  float* tbuf = (float*)(ws + off_t);
  int*   idx1 = (int*)(ws + off_idx1);
  int*   idx2 = (int*)(ws + off_idx2);
  float* xT   = (float*)(ws + off_xT);
  float* xc   = (float*)(ws + off_xc);   // rows 0..31 = x1, rows 32..543 = x2
  float* x3   = (float*)(ws + off_x3);
  float* f1   = (float*)(ws + off_f1);
  float* z1a  = (float*)(ws + off_z1a);
  float* f2   = (float*)(ws + off_f2);
  float* z2a  = (float*)(ws + off_z2a);
  float* z3a  = (float*)(ws + off_z3a);
  float* Hb   = (float*)(ws + off_H);
  float* H2   = (float*)(ws + off_H2);

  auto g1 = [](int n) { return dim3((n + 255) / 256); };
  const float* NF = nullptr;
  float* NFW = nullptr;
  int* NI = nullptr;

  // ---- point cloud transpose + knn1 + graph feature 1 ----
  k_transpose_xy<<<g1(NT), 256, 0, stream>>>(xy, xT, NT);
  k_knn<2><<<dim3(N / 256, B), 256, 0, stream>>>(xT, NT, idx1, N);
  k_feature<2><<<g1(CE), 256, 0, stream>>>(xT, NT, idx1, f1, CE, N);

  // ---- conv1 L1: GEMM(4->11) store+stats, finalize, BN+GELU ----
  k_fill_f32<<<g1(2 * SMAXCH), 256, 0, stream>>>(sums, 0.f, 2 * SMAXCH);
  k_gemm<<<dim3(1, CE / 32), 256, 0, stream>>>(W1a, b1a, f1, 11, 4, CE,
      z1a, sums, NF, NF, NF, NF, NI, 0, 1, 1 | 2);
  k_finalize_stats<<<g1(11), 256, 0, stream>>>(sums, mean, rstd, 11, 1.f / CE);
  k_bngelu<<<dim3((CE + 255) / 256, 11), 256, 0, stream>>>(z1a, mean, rstd, g1a, bt1a, CE);

  // ---- conv1 L2: GEMM(11->32) stats pass, then BN+GELU+max -> x1 (xc rows 0..31) ----
  k_fill_f32<<<g1(2 * SMAXCH), 256, 0, stream>>>(sums, 0.f, 2 * SMAXCH);
  k_gemm<<<dim3(1, CE / 32), 256, 0, stream>>>(W1b, b1b, z1a, 32, 11, CE,
      NFW, sums, NF, NF, NF, NF, NI, 0, 1, 2);
  k_finalize_stats<<<g1(32), 256, 0, stream>>>(sums, mean, rstd, 32, 1.f / CE);
  k_fill_i32<<<g1(32 * NT), 256, 0, stream>>>((int*)xc, NEGINF_KEY, 32 * NT);
  k_gemm<<<dim3(1, CE / 32), 256, 0, stream>>>(W1b, b1b, z1a, 32, 11, CE,
      NFW, NFW, mean, rstd, g1b, bt1b, (int*)xc, NT, KNN, 4);
  k_decode_max<<<g1(32 * NT), 256, 0, stream>>>(xc, 32 * NT);

  // ---- knn2 (32-dim) + graph feature 2 ----
  k_knn<32><<<dim3(N / 256, B), 256, 0, stream>>>(xc, NT, idx2, N);
  k_feature<32><<<g1(CE), 256, 0, stream>>>(xc, NT, idx2, f2, CE, N);

  // ---- conv2 L1: GEMM(64->181) store+stats, finalize, BN+GELU ----
  k_fill_f32<<<g1(2 * SMAXCH), 256, 0, stream>>>(sums, 0.f, 2 * SMAXCH);
  k_gemm<<<dim3(3, CE / 32), 256, 0, stream>>>(W2a, b2a, f2, 181, 64, CE,
      z2a, sums, NF, NF, NF, NF, NI, 0, 1, 1 | 2);
  k_finalize_stats<<<g1(181), 256, 0, stream>>>(sums, mean, rstd, 181, 1.f / CE);
  k_bngelu<<<dim3((CE + 255) / 256, 181), 256, 0, stream>>>(z2a, mean, rstd, g2a, bt2a, CE);

  // ---- conv2 L2: GEMM(181->512) stats, then BN+GELU+max -> x2 (xc rows 32..543) ----
  k_fill_f32<<<g1(2 * SMAXCH), 256, 0, stream>>>(sums, 0.f, 2 * SMAXCH);
  k_gemm<<<dim3(8, CE / 32), 256, 0, stream>>>(W2b, b2b, z2a, 512, 181, CE,
      NFW, sums, NF, NF, NF, NF, NI, 0, 1, 2);
  k_finalize_stats<<<g1(512), 256, 0, stream>>>(sums, mean, rstd, 512, 1.f / CE);
  k_fill_i32<<<g1(512 * NT), 256, 0, stream>>>((int*)(xc + 32 * NT), NEGINF_KEY, 512 * NT);
  k_gemm<<<dim3(8, CE / 32), 256, 0, stream>>>(W2b, b2b, z2a, 512, 181, CE,
      NFW, NFW, mean, rstd, g2b, bt2b, (int*)(xc + 32 * NT), NT, KNN, 4);
  k_decode_max<<<g1(512 * NT), 256, 0, stream>>>(xc + 32 * NT, 512 * NT);

  // ---- conv3 L1: GEMM(544->527) on xc, store+stats, BN+GELU ----
  k_fill_f32<<<g1(2 * SMAXCH), 256, 0, stream>>>(sums, 0.f, 2 * SMAXCH);
  k_gemm<<<dim3(9, NT / 32), 256, 0, stream>>>(W3a, b3a, xc, 527, 544, NT,
      z3a, sums, NF, NF, NF, NF, NI, 0, 1, 1 | 2);
  k_finalize_stats<<<g1(527), 256, 0, stream>>>(sums, mean, rstd, 527, 1.f / NT);
  k_bngelu<<<dim3((NT + 255) / 256, 527), 256, 0, stream>>>(z3a, mean, rstd, g3a, bt3a, NT);

  // ---- conv3 L2: GEMM(527->512) -> x3, store+stats, BN+GELU ----
  k_fill_f32<<<g1(2 * SMAXCH), 256, 0, stream>>>(sums, 0.f, 2 * SMAXCH);
  k_gemm<<<dim3(8, NT / 32), 256, 0, stream>>>(W3b, b3b, z3a, 512, 527, NT,
      x3, sums, NF, NF, NF, NF, NI, 0, 1, 1 | 2);
  k_finalize_stats<<<g1(512), 256, 0, stream>>>(sums, mean, rstd, 512, 1.f / NT);
  k_bngelu<<<dim3((NT + 255) / 256, 512), 256, 0, stream>>>(x3, mean, rstd, g3b, bt3b, NT);

  // ---- time embedding MLP ----
  k_time<<<g1(B * 1024), 256, 0, stream>>>(temb, tW, tb, tbuf, B, 1024, 256);

  // ---- block1: GEMM 512x512 + bias -> H; rmsnorm + modulate + silu in place ----
  k_gemm<<<dim3(8, NT / 32), 256, 0, stream>>>(B1W, B1b, x3, 512, 512, NT,
      Hb, NFW, NF, NF, NF, NF, NI, 0, 1, 1);
  k_blocknorm<<<dim3(NT / 16), 256, 0, stream>>>(Hb, Hb, B1g, tbuf, NF, 1, 512, NT, N, 0);

  // ---- block2: GEMM -> H2; rmsnorm + silu + residual(x3) -> d_out (batch-major) ----
  k_gemm<<<dim3(8, NT / 32), 256, 0, stream>>>(B2W, B2b, Hb, 512, 512, NT,
      H2, NFW, NF, NF, NF, NF, NI, 0, 1, 1);
  k_blocknorm<<<dim3(NT / 16), 256, 0, stream>>>(H2, (float*)d_out, B2g, NF, x3, 0, 512, NT, N, 1);
}